// NCutLoss_18425409699881
// MI455X (gfx1250) — compile-verified
//
#include <hip/hip_runtime.h>
#include <hip/hip_bf16.h>
#include <stdint.h>

#define N_NODES 2048
#define KDIM    64
#define NBINS   16
#define MPAD    2080            // 2049 k-values padded to 65 chunks of 32
#define KCHUNK  32
#define NCHUNK  (MPAD / KCHUNK) // 65
#define KPAIRS  (MPAD / 2)      // 1040
#define PROW    128             // floats per k-pair row in global P (64 l * 2)
#define LROW    160             // floats per k-pair row in LDS (padded vs bank conflicts)

typedef __attribute__((ext_vector_type(2))) float v2f;
typedef __attribute__((ext_vector_type(8))) float v8f;

__device__ __forceinline__ unsigned lds_off(const void* p) {
    // generic pointer to LDS: addr[31:0] is the LDS byte offset (ISA 10.2)
    return (unsigned)(uintptr_t)p;
}

// ---------------------------------------------------------------------------
// 1) alpha_bars = per-bin mean of probs; also log z / log(1-z) tables.
// ---------------------------------------------------------------------------
__global__ void k_alpha(const float* __restrict__ probs,
                        const int*   __restrict__ n2b,
                        const float* __restrict__ bin_counts,
                        float* __restrict__ alpha_out,   // d_out + 1
                        float* __restrict__ logz,
                        float* __restrict__ log1mz) {
    const int t = threadIdx.x;         // 0..1023
    const int b = t >> 6;
    const int l = t & 63;
    float s = 0.0f;
    for (int n = 0; n < N_NODES; ++n) {
        if (n2b[n] == b) s += probs[n * KDIM + l];
    }
    const float alpha = s / bin_counts[b];
    alpha_out[t] = alpha;
    const float z = fminf(fmaxf(alpha, 1e-7f), 1.0f - 1e-7f);
    logz[t]   = logf(z);
    log1mz[t] = log1pf(-z);
}

// ---------------------------------------------------------------------------
// 2) log binomial coefficients
// ---------------------------------------------------------------------------
__global__ void k_lbinom(float* __restrict__ lb) {
    const int k = blockIdx.x * 256 + threadIdx.x;
    if (k < MPAD) {
        float v = 0.0f;
        if (k <= N_NODES) {
            const float kf = (float)k;
            v = lgammaf((float)N_NODES + 1.0f)
              - lgammaf(kf + 1.0f)
              - lgammaf((float)N_NODES - kf + 1.0f);
        }
        lb[k] = v;
    }
}

// ---------------------------------------------------------------------------
// 3) pmf table, k-paired layout: P[b][k>>1][l][k&1]; zero pad for k > 2048.
// ---------------------------------------------------------------------------
__global__ void k_pmf(const float* __restrict__ logz,
                      const float* __restrict__ log1mz,
                      const float* __restrict__ lbinom,
                      float* __restrict__ Ptab) {
    const int bk = blockIdx.x;          // b*MPAD + k
    const int l  = threadIdx.x;
    const int b  = bk / MPAD;
    const int k  = bk - b * MPAD;
    float val = 0.0f;
    if (k <= N_NODES) {
        const float kf = (float)k;
        const float lp = lbinom[k]
                       + kf * logz[b * KDIM + l]
                       + ((float)N_NODES - kf) * log1mz[b * KDIM + l];
        val = expf(lp);   // far-tail underflow to 0 matches f32 reference
    }
    Ptab[(size_t)b * KPAIRS * PROW + (size_t)(k >> 1) * PROW + l * 2 + (k & 1)] = val;
}

// ---------------------------------------------------------------------------
// 4) Batched WMMA GEMM: f_val[v][b][l] = sum_k (cm1/(cm1+k)) * P[b][k][l]
//    Grid (N/128, NBINS), 256 threads (8 waves, 16 rows x 64 cols each).
//    B staged per-chunk into LDS via global_load_async_to_lds_b128 (double
//    buffered, ASYNCcnt), read back as ds_load_b64; A generated with v_rcp_f32.
// ---------------------------------------------------------------------------
__global__ void __launch_bounds__(256)
k_hyp(const float* __restrict__ deg,
      const float* __restrict__ beta,
      const float* __restrict__ Ptab,
      float* __restrict__ Ftab) {
    __shared__ float buf[2][16 * LROW];          // 2 x 10240 B

    const int b    = blockIdx.y;
    const int tid  = threadIdx.x;
    const int wave = tid >> 5;
    const int lane = tid & 31;
    const int v0   = blockIdx.x * 128 + wave * 16;
    const int mrow = lane & 15;
    const int hsel = lane >> 4;                  // 0: K={0,1}, 1: K={2,3}

    const float q   = fmaxf(deg[v0 + mrow], 1e-6f);
    const float cm1 = q / beta[b];
    const float* Pb = Ptab + (size_t)b * KPAIRS * PROW;

    // async-copy slices: 512 x 16B per chunk, 2 per thread
    const unsigned row0 = (unsigned)tid >> 5,          seg0 = (unsigned)tid & 31;
    const unsigned row1 = (unsigned)(tid + 256) >> 5,  seg1 = (unsigned)(tid + 256) & 31;
    const unsigned d0[2] = { lds_off(&buf[0][row0 * LROW]) + seg0 * 16,
                             lds_off(&buf[1][row0 * LROW]) + seg0 * 16 };
    const unsigned d1[2] = { lds_off(&buf[0][row1 * LROW]) + seg1 * 16,
                             lds_off(&buf[1][row1 * LROW]) + seg1 * 16 };
    const unsigned g0 = row0 * 512 + seg0 * 16;  // byte offset inside a chunk
    const unsigned g1 = row1 * 512 + seg1 * 16;

    // prologue: stage chunk 0 into buffer 0
    asm volatile("global_load_async_to_lds_b128 %0, %1, %2"
                 :: "v"(d0[0]), "v"(g0), "s"(Pb) : "memory");
    asm volatile("global_load_async_to_lds_b128 %0, %1, %2"
                 :: "v"(d1[0]), "v"(g1), "s"(Pb) : "memory");
    asm volatile("s_wait_asynccnt 0" ::: "memory");
    __syncthreads();

    v8f acc0 = {}, acc1 = {}, acc2 = {}, acc3 = {};
    float kf = (float)(hsel << 1);               // running k for this half-wave

    for (int c = 0; c < NCHUNK; ++c) {
        const int cur = c & 1;
        if (c + 1 < NCHUNK) {                    // prefetch next chunk (overlaps WMMA)
            const unsigned off = (unsigned)(c + 1) * (KCHUNK / 2) * PROW * 4;
            asm volatile("global_load_async_to_lds_b128 %0, %1, %2"
                         :: "v"(d0[cur ^ 1]), "v"(g0 + off), "s"(Pb) : "memory");
            asm volatile("global_load_async_to_lds_b128 %0, %1, %2"
                         :: "v"(d1[cur ^ 1]), "v"(g1 + off), "s"(Pb) : "memory");
        }
        const float* bp = &buf[cur][0];
        #pragma unroll
        for (int ks = 0; ks < KCHUNK; ks += 4) {
            v2f a;
            a.x = cm1 * __builtin_amdgcn_rcpf(cm1 + kf);
            a.y = cm1 * __builtin_amdgcn_rcpf(cm1 + kf + 1.0f);
            kf += 4.0f;

            const float* r = bp + ((ks >> 1) + hsel) * LROW + mrow * 2;
            const v2f b0 = *(const v2f*)(r);
            const v2f b1 = *(const v2f*)(r + 32);
            const v2f b2 = *(const v2f*)(r + 64);
            const v2f b3 = *(const v2f*)(r + 96);

            acc0 = __builtin_amdgcn_wmma_f32_16x16x4_f32(false, a, false, b0, (short)0, acc0, false, false);
            acc1 = __builtin_amdgcn_wmma_f32_16x16x4_f32(false, a, false, b1, (short)0, acc1, false, false);
            acc2 = __builtin_amdgcn_wmma_f32_16x16x4_f32(false, a, false, b2, (short)0, acc2, false, false);
            acc3 = __builtin_amdgcn_wmma_f32_16x16x4_f32(false, a, false, b3, (short)0, acc3, false, false);
        }
        asm volatile("s_wait_asynccnt 0" ::: "memory");
        __syncthreads();
    }

    const int rhi = hsel * 8;                    // C layout: lanes 16-31 -> M=r+8
    const int col = lane & 15;
    for (int r = 0; r < 8; ++r) {
        const int vr = v0 + r + rhi;
        float* o = Ftab + ((size_t)vr * NBINS + b) * KDIM;
        o[col]      = acc0[r];
        o[16 + col] = acc1[r];
        o[32 + col] = acc2[r];
        o[48 + col] = acc3[r];
    }
}

// ---------------------------------------------------------------------------
// 5) phi[v][l] = exp( sum_b bw[b] * log(max(f_val[v][b][l]/q[v], 1e-30)) )
// ---------------------------------------------------------------------------
__global__ void k_phi(const float* __restrict__ deg,
                      const float* __restrict__ bw,
                      const float* __restrict__ Ftab,
                      float* __restrict__ phi) {
    const int idx = blockIdx.x * 256 + threadIdx.x;   // v*64 + l
    const int v = idx >> 6;
    const int l = idx & 63;
    const float rq = 1.0f / fmaxf(deg[v], 1e-6f);
    const float* Fv = Ftab + (size_t)v * (NBINS * KDIM);
    float acc = 0.0f;
    #pragma unroll
    for (int b = 0; b < NBINS; ++b) {
        acc += bw[b] * logf(fmaxf(Fv[b * KDIM + l] * rq, 1e-30f));
    }
    phi[idx] = expf(acc);
}

// ---------------------------------------------------------------------------
// 6a) Bt[kpair][l][2] = 1 - probs[k][l]  (paired layout for b64 B loads)
// ---------------------------------------------------------------------------
__global__ void k_prep(const float* __restrict__ probs, float* __restrict__ Bt) {
    const int idx = blockIdx.x * 256 + threadIdx.x;   // k*64 + l
    const int k = idx >> 6;
    const int l = idx & 63;
    Bt[(size_t)(k >> 1) * PROW + l * 2 + (k & 1)] = 1.0f - probs[idx];
}

// ---------------------------------------------------------------------------
// 6b) WMMA GEMM: G = w_mat @ (1 - probs)   (2048 x 2048 x 64, f32)
// ---------------------------------------------------------------------------
__global__ void __launch_bounds__(128)
k_gemm2(const float* __restrict__ w,
        const float* __restrict__ Bt,
        float* __restrict__ G) {
    const int wave = threadIdx.x >> 5;
    const int lane = threadIdx.x & 31;
    const int v0   = blockIdx.x * 64 + wave * 16;
    const int mrow = lane & 15;
    const int hsel = lane >> 4;

    const float* wrow = w + (size_t)(v0 + mrow) * N_NODES + (hsel << 1);
    v8f acc0 = {}, acc1 = {}, acc2 = {}, acc3 = {};
    for (int kc = 0; kc < N_NODES; kc += 4) {
        const v2f a = *(const v2f*)(wrow + kc);
        const float* r = Bt + (size_t)((kc >> 1) + hsel) * PROW + mrow * 2;
        const v2f b0 = *(const v2f*)(r);
        const v2f b1 = *(const v2f*)(r + 32);
        const v2f b2 = *(const v2f*)(r + 64);
        const v2f b3 = *(const v2f*)(r + 96);

        acc0 = __builtin_amdgcn_wmma_f32_16x16x4_f32(false, a, false, b0, (short)0, acc0, false, false);
        acc1 = __builtin_amdgcn_wmma_f32_16x16x4_f32(false, a, false, b1, (short)0, acc1, false, false);
        acc2 = __builtin_amdgcn_wmma_f32_16x16x4_f32(false, a, false, b2, (short)0, acc2, false, false);
        acc3 = __builtin_amdgcn_wmma_f32_16x16x4_f32(false, a, false, b3, (short)0, acc3, false, false);
    }

    const int rhi = hsel * 8;
    const int col = lane & 15;
    for (int r = 0; r < 8; ++r) {
        const int vr = v0 + r + rhi;
        float* o = G + (size_t)vr * KDIM;
        o[col]      = acc0[r];
        o[16 + col] = acc1[r];
        o[32 + col] = acc2[r];
        o[48 + col] = acc3[r];
    }
}

// ---------------------------------------------------------------------------
// 7) Partial sums of w_mat (deterministic two-stage reduction)
// ---------------------------------------------------------------------------
__global__ void k_wsum(const float* __restrict__ w, float* __restrict__ partial) {
    __shared__ float sm[256];
    float s = 0.0f;
    const size_t total = (size_t)N_NODES * N_NODES;
    for (size_t i = (size_t)blockIdx.x * 256 + threadIdx.x; i < total; i += (size_t)256 * 1024)
        s += w[i];
    sm[threadIdx.x] = s;
    __syncthreads();
    for (int o = 128; o > 0; o >>= 1) {
        if (threadIdx.x < o) sm[threadIdx.x] += sm[threadIdx.x + o];
        __syncthreads();
    }
    if (threadIdx.x == 0) partial[blockIdx.x] = sm[0];
}

// ---------------------------------------------------------------------------
// 8) Final: wsum = sum(partial); num = sum(probs*G*phi); loss = num/max(wsum,1e-9)
// ---------------------------------------------------------------------------
__global__ void k_final(const float* __restrict__ partial,
                        const float* __restrict__ probs,
                        const float* __restrict__ G,
                        const float* __restrict__ phi,
                        float* __restrict__ out) {
    __shared__ float sm[256];
    const int t = threadIdx.x;

    float s = 0.0f;
    for (int i = t; i < 1024; i += 256) s += partial[i];
    sm[t] = s;
    __syncthreads();
    for (int o = 128; o > 0; o >>= 1) {
        if (t < o) sm[t] += sm[t + o];
        __syncthreads();
    }
    const float wsum = sm[0];
    __syncthreads();

    float num = 0.0f;
    for (int i = t; i < N_NODES * KDIM; i += 256)
        num += probs[i] * G[i] * phi[i];
    sm[t] = num;
    __syncthreads();
    for (int o = 128; o > 0; o >>= 1) {
        if (t < o) sm[t] += sm[t + o];
        __syncthreads();
    }
    if (t == 0) out[0] = sm[0] / fmaxf(wsum, 1e-9f);
}

// ---------------------------------------------------------------------------
// Host launcher
// ---------------------------------------------------------------------------
extern "C" void kernel_launch(void* const* d_in, const int* in_sizes, int n_in,
                              void* d_out, int out_size, void* d_ws, size_t ws_size,
                              hipStream_t stream) {
    const float* w     = (const float*)d_in[0];   // (N,N)
    const float* probs = (const float*)d_in[1];   // (N,K)
    const float* deg   = (const float*)d_in[2];   // (N)
    const float* beta  = (const float*)d_in[3];   // (B)
    const float* bw    = (const float*)d_in[4];   // (B)
    const float* bc    = (const float*)d_in[5];   // (B)
    const int*   n2b   = (const int*)d_in[6];     // (N) int32
    float* out = (float*)d_out;                   // [loss, alpha_bars(16*64)]
    float* ws  = (float*)d_ws;

    // workspace layout (floats)
    float* logz    = ws;                                   // 1024
    float* log1mz  = ws + 1024;                            // 1024
    float* lbinom  = ws + 2048;                            // 2080
    float* Ptab    = ws + 4128;                            // 16*1040*128 (16B-aligned)
    float* Ftab    = Ptab + (size_t)NBINS * KPAIRS * PROW; // 2048*16*64
    float* phi     = Ftab + (size_t)N_NODES * NBINS * KDIM;// 2048*64
    float* Bt      = phi + N_NODES * KDIM;                 // 1024*128
    float* G       = Bt + (size_t)(N_NODES / 2) * PROW;    // 2048*64
    float* partial = G + N_NODES * KDIM;                   // 1024

    k_alpha <<<1, 1024, 0, stream>>>(probs, n2b, bc, out + 1, logz, log1mz);
    k_lbinom<<<(MPAD + 255) / 256, 256, 0, stream>>>(lbinom);
    k_pmf   <<<NBINS * MPAD, 64, 0, stream>>>(logz, log1mz, lbinom, Ptab);
    k_hyp   <<<dim3(N_NODES / 128, NBINS), 256, 0, stream>>>(deg, beta, Ptab, Ftab);
    k_phi   <<<(N_NODES * KDIM) / 256, 256, 0, stream>>>(deg, bw, Ftab, phi);
    k_prep  <<<(N_NODES * KDIM) / 256, 256, 0, stream>>>(probs, Bt);
    k_gemm2 <<<N_NODES / 64, 128, 0, stream>>>(w, Bt, G);
    k_wsum  <<<1024, 256, 0, stream>>>(w, partial);
    k_final <<<1, 256, 0, stream>>>(partial, probs, G, phi, out);
}